// DetectionNetwork_13134009991803
// MI455X (gfx1250) — compile-verified
//
#include <hip/hip_runtime.h>
#include <hip/hip_bf16.h>

// ---------------- constants ----------------
#define BATCH 2
#define PRE_K 1000
#define PREP  1008          // padded to 63*16 for WMMA tiles
#define POST_K 100
#define CAPC  16384
#define NBINS 4096
#define NTIL  63            // 63*16 = 1008
#define TWO_PI_F 6.28318530717958647692f
#define THR_SCORE_F 0.3f
#define THR_IOU_F 0.01f

// ---------------- workspace layout (uint32 words, per-batch stride WSB) ----
constexpr int OFF_HIST = 0;                       // NBINS
constexpr int OFF_CNT  = OFF_HIST + NBINS;        // 1
constexpr int OFF_THR  = OFF_CNT + 1;             // 1
constexpr int OFF_CAND = OFF_THR + 1;             // 2*CAPC (key, idx) -- 8B aligned
constexpr int OFF_BOX  = OFF_CAND + 2*CAPC;       // 7*PRE_K floats
constexpr int OFF_SC   = OFF_BOX + 7*PRE_K;       // PRE_K floats
constexpr int OFF_DL   = OFF_SC + PRE_K;          // PRE_K u32
constexpr int OFF_VAL  = OFF_DL + PRE_K;          // PRE_K u32
constexpr int OFF_PX   = OFF_VAL + PRE_K;         // PREP floats (centers x)
constexpr int OFF_PY   = OFF_PX + PREP;           // PREP floats
constexpr int OFF_PW   = OFF_PY + PREP;           // PREP floats (w)
constexpr int OFF_PL   = OFF_PW + PREP;           // PREP floats (l)
constexpr int OFF_PR   = OFF_PL + PREP;           // PREP floats (rot)
constexpr int OFF_PN   = OFF_PR + PREP;           // PREP floats (x^2+y^2)
constexpr int OFF_PHD  = OFF_PN + PREP;           // PREP floats (half diag)
constexpr int OFF_ADJ  = OFF_PHD + PREP;          // 32*PRE_K adjacency bitmask
constexpr int WSB      = OFF_ADJ + 32*PRE_K;

typedef __attribute__((ext_vector_type(2))) float v2f;
typedef __attribute__((ext_vector_type(8))) float v8f;

// monotone float -> uint key (ascending)
__device__ __forceinline__ unsigned fkey(float f) {
    unsigned u = __float_as_uint(f);
    return (u & 0x80000000u) ? ~u : (u | 0x80000000u);
}

// ---------------- K0: zero workspace ----------------
__global__ void k_zero(unsigned* ws) {
    const long total = (long)BATCH * WSB;
    for (long i = blockIdx.x * (long)blockDim.x + threadIdx.x; i < total;
         i += (long)gridDim.x * blockDim.x)
        ws[i] = 0u;
}

// ---------------- K1: score histogram (streaming, prefetch) ----------------
__global__ void k_hist(const float* __restrict__ cls, unsigned* ws, int N) {
    const long total = (long)BATCH * N;
    const long stride = (long)gridDim.x * blockDim.x;
    for (long i = blockIdx.x * (long)blockDim.x + threadIdx.x; i < total; i += stride) {
        __builtin_prefetch(cls + i + stride, 0, 0);   // global_prefetch_b8
        int b = (i >= N) ? 1 : 0;
        unsigned key = fkey(cls[i]);
        atomicAdd(ws + (size_t)b*WSB + OFF_HIST + (key >> 20), 1u);
    }
}

// ---------------- K2: pick threshold bin ----------------
__global__ void k_thr(unsigned* ws) {
    unsigned* W = ws + (size_t)blockIdx.x * WSB;
    unsigned acc = 0; int sel = 0;
    for (int k = NBINS - 1; k >= 0; --k) {
        acc += W[OFF_HIST + k];
        if (acc >= PRE_K) { sel = k; break; }
    }
    W[OFF_THR] = (unsigned)sel;
}

// ---------------- K3: compact candidates above threshold ----------------
__global__ void k_collect(const float* __restrict__ cls, unsigned* ws, int N) {
    const long total = (long)BATCH * N;
    const long stride = (long)gridDim.x * blockDim.x;
    for (long i = blockIdx.x * (long)blockDim.x + threadIdx.x; i < total; i += stride) {
        __builtin_prefetch(cls + i + stride, 0, 0);
        int b = (i >= N) ? 1 : 0;
        unsigned idx = (unsigned)(i - (long)b * N);
        unsigned key = fkey(cls[i]);
        unsigned* W = ws + (size_t)b*WSB;
        if ((key >> 20) >= W[OFF_THR]) {
            unsigned pos = atomicAdd(W + OFF_CNT, 1u);
            if (pos < CAPC) { W[OFF_CAND + 2*pos] = key; W[OFF_CAND + 2*pos + 1] = idx; }
        }
    }
}

// ---------------- K4: rank candidates (stable top-k), gather-decode -------
// Candidate tiles are staged global->LDS with GLOBAL_LOAD_ASYNC_TO_LDS_B64
// (CDNA5 async path, ASYNCcnt), drained with s_wait_asynccnt.
__global__ __launch_bounds__(256) void k_select(const float* __restrict__ anchors,
                                                const float* __restrict__ boxp,
                                                const float* __restrict__ cls,
                                                const float* __restrict__ dirp,
                                                unsigned* ws, int N) {
    const int b = blockIdx.y;
    unsigned* W = ws + (size_t)b*WSB;
    float* Wf = (float*)W;
    unsigned M = W[OFF_CNT]; if (M > CAPC) M = CAPC;

    __shared__ __align__(16) unsigned spair[2048];   // 1024 (key,idx) pairs = 8KB
    const int m = blockIdx.x * blockDim.x + threadIdx.x;
    const bool active = (unsigned)m < M;
    unsigned key = 0, idx = 0;
    if (active) { key = W[OFF_CAND + 2*m]; idx = W[OFF_CAND + 2*m + 1]; }

    unsigned rank = 0;
    for (unsigned base = 0; base < M; base += 1024) {
        unsigned cnt = M - base; if (cnt > 1024) cnt = 1024;
        // async-copy the full 1024-pair tile (CAPC is a multiple of 1024, so
        // reads past M stay inside the candidate buffer; rank loop bounds cnt)
#pragma unroll
        for (int q = 0; q < 4; ++q) {
            unsigned t = threadIdx.x + (unsigned)q * 256u;
            unsigned ldsoff = (unsigned)(unsigned long long)(&spair[2*t]);
            unsigned long long ga =
                (unsigned long long)(W + OFF_CAND + 2*(base + t));
            asm volatile("global_load_async_to_lds_b64 %0, %1, off"
                         :: "v"(ldsoff), "v"(ga) : "memory");
        }
        asm volatile("s_wait_asynccnt 0x0" ::: "memory");
        __syncthreads();
        if (active) {
            for (unsigned k = 0; k < cnt; ++k) {
                unsigned kk = spair[2*k];
                rank += (kk > key) || (kk == key && spair[2*k + 1] < idx);
            }
        }
        __syncthreads();
    }
    if (!active || rank >= PRE_K) return;

    const long base = ((long)b * N + idx);
    const float* a = anchors + base * 7;
    const float* t = boxp + base * 7;
    float xa=a[0], ya=a[1], za=a[2], wa=a[3], la=a[4], ha=a[5], ra=a[6];
    float xt=t[0], yt=t[1], zt=t[2], wt=t[3], lt=t[4], ht=t[5], rt=t[6];
    za += ha * 0.5f;
    float diag = sqrtf(la*la + wa*wa);
    float xg = xt*diag + xa, yg = yt*diag + ya, zg = zt*ha + za;
    float lg = expf(lt)*la, wg = expf(wt)*wa, hg = expf(ht)*ha;
    float rg = rt + ra;
    zg -= hg * 0.5f;

    const int s = (int)rank;
    Wf[OFF_BOX + s*7 + 0] = xg; Wf[OFF_BOX + s*7 + 1] = yg; Wf[OFF_BOX + s*7 + 2] = zg;
    Wf[OFF_BOX + s*7 + 3] = wg; Wf[OFF_BOX + s*7 + 4] = lg; Wf[OFF_BOX + s*7 + 5] = hg;
    Wf[OFF_BOX + s*7 + 6] = rg;

    float logit = cls[base];
    float score = 1.0f / (1.0f + expf(-logit));
    Wf[OFF_SC + s] = score;
    const float* dd = dirp + base * 2;
    W[OFF_DL + s] = (dd[1] > dd[0]) ? 1u : 0u;
    W[OFF_VAL + s] = (score > THR_SCORE_F) ? 1u : 0u;
    Wf[OFF_PX + s] = xg; Wf[OFF_PY + s] = yg;
    Wf[OFF_PW + s] = wg; Wf[OFF_PL + s] = lg; Wf[OFF_PR + s] = rg;
    Wf[OFF_PN + s] = xg*xg + yg*yg;
    Wf[OFF_PHD + s] = 0.5f * sqrtf(wg*wg + lg*lg);
}

// ---------------- exact rotated rectangle intersection (Sutherland–Hodgman)
__device__ int clip_half(const float* xi, const float* yi, int n,
                         float* xo, float* yo, float A, float Bc, float D) {
    int m = 0;
    for (int k = 0; k < n; ++k) {
        int k2 = (k + 1 == n) ? 0 : k + 1;
        float f1 = A*xi[k]  + Bc*yi[k]  - D;
        float f2 = A*xi[k2] + Bc*yi[k2] - D;
        if (f1 <= 0.f) { xo[m] = xi[k]; yo[m] = yi[k]; ++m; }
        if ((f1 < 0.f) != (f2 < 0.f)) {
            float tt = f1 / (f1 - f2);
            xo[m] = xi[k] + tt * (xi[k2] - xi[k]);
            yo[m] = yi[k] + tt * (yi[k2] - yi[k]);
            ++m;
        }
    }
    return m;
}

__device__ float rect_inter_area(float ax, float ay, float aw, float al, float ar,
                                 float bx, float by, float bw, float bl, float br) {
    float ca = cosf(ar), sa = sinf(ar);
    float cb = cosf(br), sb = sinf(br);
    float hw = 0.5f*aw, hl = 0.5f*al;
    float qx[8], qy[8], tx[8], ty[8];
    const float dxs[4] = { hw,  hw, -hw, -hw };
    const float dys[4] = { hl, -hl, -hl,  hl };
#pragma unroll
    for (int k = 0; k < 4; ++k) {
        float wx = ax + dxs[k]*ca - dys[k]*sa;
        float wy = ay + dxs[k]*sa + dys[k]*ca;
        float dx = wx - bx, dy = wy - by;
        qx[k] =  dx*cb + dy*sb;      // A corner in B's local frame
        qy[k] = -dx*sb + dy*cb;
    }
    float HBW = 0.5f*bw, HBL = 0.5f*bl;
    int n = 4;
    n = clip_half(qx, qy, n, tx, ty,  1.f, 0.f, HBW);
    n = clip_half(tx, ty, n, qx, qy, -1.f, 0.f, HBW);
    n = clip_half(qx, qy, n, tx, ty,  0.f, 1.f, HBL);
    n = clip_half(tx, ty, n, qx, qy,  0.f,-1.f, HBL);
    if (n < 3) return 0.f;
    float area = 0.f;
    for (int k = 0; k < n; ++k) {
        int k2 = (k + 1 == n) ? 0 : k + 1;
        area += qx[k]*qy[k2] - qx[k2]*qy[k];
    }
    return 0.5f * fabsf(area);
}

// ---------------- K5: WMMA center-Gram prefilter + exact IoU → adjacency --
__global__ __launch_bounds__(256) void k_iou(unsigned* ws) {
    const int b = blockIdx.y;
    unsigned* W = ws + (size_t)b*WSB;
    float* Wf = (float*)W;
    const int wave = threadIdx.x >> 5;
    const int lane = threadIdx.x & 31;
    const int tile = blockIdx.x * 8 + wave;
    if (tile >= NTIL*NTIL) return;
    const int ti = tile / NTIL, tj = tile % NTIL;
    const int i0 = ti * 16, j0 = tj * 16;

    const float* px  = Wf + OFF_PX;  const float* py  = Wf + OFF_PY;
    const float* pw  = Wf + OFF_PW;  const float* pl  = Wf + OFF_PL;
    const float* pr  = Wf + OFF_PR;  const float* pn  = Wf + OFF_PN;
    const float* phd = Wf + OFF_PHD;

    // A: 16x4 (K=0,1 = x,y in lanes 0..15; K=2,3 zero in lanes 16..31)
    v2f av, bv;
    if (lane < 16) { av.x = px[i0 + lane]; av.y = py[i0 + lane]; }
    else           { av.x = 0.f;           av.y = 0.f; }
    if (lane < 16) { bv.x = px[j0 + lane]; bv.y = py[j0 + lane]; }
    else           { bv.x = 0.f;           bv.y = 0.f; }
    v8f cacc = {0.f,0.f,0.f,0.f,0.f,0.f,0.f,0.f};
    // D = A x B : G[i][j] = x_i*x_j + y_i*y_j
    v8f g = __builtin_amdgcn_wmma_f32_16x16x4_f32(
        false, av, false, bv, (short)0, cacc, false, false);

#pragma unroll
    for (int v = 0; v < 8; ++v) {
        int i = i0 + v + ((lane >> 4) << 3);   // C/D layout: lanes 16..31 -> M+8
        int j = j0 + (lane & 15);
        if (i >= PRE_K || j >= PRE_K || i == j) continue;
        float d2 = pn[i] + pn[j] - 2.f * g[v];
        float rthr = phd[i] + phd[j] + 0.25f;  // margin for fp rounding
        if (d2 > rthr * rthr) continue;        // prefilter: cannot overlap
        float inter = rect_inter_area(px[i], py[i], pw[i], pl[i], pr[i],
                                      px[j], py[j], pw[j], pl[j], pr[j]);
        float areaI = pw[i]*pl[i], areaJ = pw[j]*pl[j];
        float iou = inter / fmaxf(areaI + areaJ - inter, 1e-8f);
        if (iou > THR_IOU_F)
            atomicOr(W + OFF_ADJ + i*32 + (j >> 5), 1u << (j & 31));
    }
}

// ---------------- K6: sequential NMS scan (1 wave per batch) + output -----
__global__ void k_nms(unsigned* ws, float* __restrict__ out) {
    const int b = blockIdx.x;
    unsigned* W = ws + (size_t)b*WSB;
    float* Wf = (float*)W;
    const int lane = threadIdx.x;   // blockDim = 32

    __shared__ unsigned supp[32];
    __shared__ unsigned vbits[32];
    __shared__ int slots[POST_K];

    supp[lane] = 0u;
    unsigned vb = 0u;
    for (int k = 0; k < 32; ++k) {
        int i = lane * 32 + k;
        if (i < PRE_K && W[OFF_VAL + i]) vb |= (1u << k);
    }
    vbits[lane] = vb;
    __syncthreads();

    int count = 0;
    for (int i = 0; i < PRE_K; ++i) {
        unsigned sb = (supp[i >> 5] >> (i & 31)) & 1u;
        unsigned vv = (vbits[i >> 5] >> (i & 31)) & 1u;
        bool keep = vv && !sb;
        if (keep) {
            supp[lane] |= W[OFF_ADJ + i*32 + lane];
            if (lane == 0 && count < POST_K) slots[count] = i;
            ++count;
        }
        __syncthreads();
    }
    int nk = count < POST_K ? count : POST_K;
    __syncthreads();

    // output layout (floats): boxes[2*100*7] | scores[200] | labels[200] | valid[200]
    for (int s = lane; s < POST_K; s += 32) {
        float bx7[7];
        float sc = 0.f, vm = 0.f;
        if (s < nk) {
            int i = slots[s];
#pragma unroll
            for (int k = 0; k < 7; ++k) bx7[k] = Wf[OFF_BOX + i*7 + k];
            float r = bx7[6];
            float dir_rot = r - floorf(r * (1.0f / TWO_PI_F) + 1.0f) * TWO_PI_F;
            bx7[6] = dir_rot + TWO_PI_F * (float)W[OFF_DL + i];
            sc = Wf[OFF_SC + i];
            vm = 1.0f;
        } else {
#pragma unroll
            for (int k = 0; k < 7; ++k) bx7[k] = 0.f;
        }
#pragma unroll
        for (int k = 0; k < 7; ++k)
            out[((long)b * POST_K + s) * 7 + k] = bx7[k];
        out[BATCH*POST_K*7 + 0*BATCH*POST_K + b*POST_K + s] = sc;   // scores
        out[BATCH*POST_K*7 + 1*BATCH*POST_K + b*POST_K + s] = 0.f;  // labels
        out[BATCH*POST_K*7 + 2*BATCH*POST_K + b*POST_K + s] = vm;   // valid
    }
}

// ---------------- launcher ----------------
extern "C" void kernel_launch(void* const* d_in, const int* in_sizes, int n_in,
                              void* d_out, int out_size, void* d_ws, size_t ws_size,
                              hipStream_t stream) {
    const float* anchors = (const float*)d_in[0];
    const float* boxp    = (const float*)d_in[1];
    const float* cls     = (const float*)d_in[2];
    const float* dirp    = (const float*)d_in[3];
    float* out = (float*)d_out;
    unsigned* ws = (unsigned*)d_ws;
    const int N = in_sizes[2] / BATCH;   // cls_preds is (B, N)

    k_zero<<<512, 256, 0, stream>>>(ws);
    k_hist<<<1024, 256, 0, stream>>>(cls, ws, N);
    k_thr<<<BATCH, 1, 0, stream>>>(ws);
    k_collect<<<1024, 256, 0, stream>>>(cls, ws, N);

    dim3 g4(CAPC / 256, BATCH);
    k_select<<<g4, 256, 0, stream>>>(anchors, boxp, cls, dirp, ws, N);

    dim3 g5((NTIL*NTIL + 7) / 8, BATCH);
    k_iou<<<g5, 256, 0, stream>>>(ws);

    k_nms<<<BATCH, 32, 0, stream>>>(ws, out);
}